// SpectralConv1d_57552561766448
// MI455X (gfx1250) — compile-verified
//
#include <hip/hip_runtime.h>

// SpectralConv1d on MI455X (gfx1250).
// out[b,o,m<512] = 0.5*s2(m) * sum_i [ x1[b,i,m]*(w[i,o,m]+w[i,o,m'])
//                                    + x1[b,i,m']*(w[i,o,m]-w[i,o,m']) ]
// out[b,o,m>=512] = 0
// Implemented as 512 independent 16x128x64 GEMMs on V_WMMA_F32_16X16X4_F32.
// Runtime is dominated by the 134 MB zero-fill (~5.8 us @ 23.3 TB/s); all
// compute inputs (2 MB x-slice + 8 MB weights) are L2-resident (192 MB L2).

typedef __attribute__((ext_vector_type(2))) float v2f;
typedef __attribute__((ext_vector_type(4))) float v4f;
typedef __attribute__((ext_vector_type(8))) float v8f;

#define B_DIM 16
#define C_DIM 64
#define L_DIM 65536
#define M_DIM 512
#define LH_DIM 32769  // L/2 + 1

#define TWO_PI 6.28318530717958647692f

// ---------------------------------------------------------------------------
// Kernel 1: zero-fill the 134 MB output with nontemporal 128-bit stores.
// ---------------------------------------------------------------------------
__global__ void __launch_bounds__(256) fill_zero_kernel(float* __restrict__ out,
                                                        long long n_elems) {
  const long long n4 = n_elems >> 2;  // out_size is a multiple of 4
  v4f z = {0.0f, 0.0f, 0.0f, 0.0f};
  v4f* p = (v4f*)out;
  long long i = (long long)blockIdx.x * blockDim.x + threadIdx.x;
  const long long stride = (long long)gridDim.x * blockDim.x;
  for (; i < n4; i += stride) {
    __builtin_nontemporal_store(z, &p[i]);
  }
}

// ---------------------------------------------------------------------------
// Kernel 2: per-position WMMA GEMM. One wave32 per spectral position m.
// Wave computes the full 16(b) x 64(o) tile for its m with K = 2 passes of 64:
//   pass0: A = x[:, :, m ]*h(m ),  B = W[:,:,m] + W[:,:,m']
//   pass1: A = x[:, :, m']*h(m'),  B = W[:,:,m] - W[:,:,m']
// ---------------------------------------------------------------------------
__global__ void __launch_bounds__(256) spectral_wmma_kernel(const float* __restrict__ x,
                                                            const float* __restrict__ w,
                                                            float* __restrict__ out) {
  const int m    = blockIdx.x * 8 + (threadIdx.x >> 5);  // 64 blocks * 8 waves = 512
  const int lane = threadIdx.x & 31;
  const int half = lane >> 4;   // 0: lanes 0-15, 1: lanes 16-31
  const int row  = lane & 15;   // A: M index (b) ; B: N index (o within tile)
  const int mp   = (M_DIM - m) & (M_DIM - 1);  // mirror position

  // theta = 2*pi*m/L : fold reciprocals into compile-time constant multiplies
  const float th  = (float)m  * (TWO_PI / (float)L_DIM);
  const float thp = (float)mp * (TWO_PI / (float)L_DIM);
  const float hm  = __builtin_cosf(th)  - __builtin_sinf(th);
  const float hmp = __builtin_cosf(thp) - __builtin_sinf(thp);

  v8f acc[4];
#pragma unroll
  for (int t = 0; t < 4; ++t) acc[t] = (v8f){0, 0, 0, 0, 0, 0, 0, 0};

#pragma unroll 1
  for (int s = 0; s < 2; ++s) {
    const int   msel = s ? mp : m;
    const float h    = s ? hmp : hm;
    const float sgn  = s ? -1.0f : 1.0f;
    // x[b, i, msel] with b = row; columns i advance along K
    const float* xb = x + (size_t)row * C_DIM * L_DIM + (size_t)msel;

    for (int k0 = 0; k0 < C_DIM; k0 += 4) {
      // A fragment (16x4 f32): v0 = K=k0+2*half, v1 = K=k0+2*half+1 per ISA layout
      const int ka = k0 + 2 * half;
      v2f a;
      a.x = xb[(size_t)ka * L_DIM] * h;
      a.y = xb[(size_t)(ka + 1) * L_DIM] * h;

#pragma unroll
      for (int t = 0; t < 4; ++t) {
        const int o = t * 16 + row;  // N index
        const float* w0 = w + ((size_t)ka * C_DIM + o) * M_DIM;
        const float* w1 = w + ((size_t)(ka + 1) * C_DIM + o) * M_DIM;
        v2f b;
        b.x = w0[m] + sgn * w0[mp];
        b.y = w1[m] + sgn * w1[mp];
        // D = A(16x4) * B(4x16) + C, fp32 throughout
        acc[t] = __builtin_amdgcn_wmma_f32_16x16x4_f32(
            /*neg_a=*/false, a, /*neg_b=*/false, b,
            /*c_mod=*/(short)0, acc[t], /*reuse_a=*/false, /*reuse_b=*/false);
      }
    }
  }

  // Final scale: 0.5 (einsum combination) * (cos-sin)(2*pi*m/Lh) / Cin.
  // Use a compile-time reciprocal so no v_div_* sequence is emitted.
  const float th2 = (float)m * (TWO_PI / (float)LH_DIM);
  const float s2 =
      (__builtin_cosf(th2) - __builtin_sinf(th2)) * (0.5f / (float)C_DIM);

  // D layout: VGPR j -> M = j + 8*half ; N = lane & 15
#pragma unroll
  for (int t = 0; t < 4; ++t) {
    const int o = t * 16 + row;
#pragma unroll
    for (int j = 0; j < 8; ++j) {
      const int b = j + 8 * half;
      out[((size_t)b * C_DIM + o) * LH_DIM + (size_t)m] = acc[t][j] * s2;
    }
  }
}

extern "C" void kernel_launch(void* const* d_in, const int* in_sizes, int n_in,
                              void* d_out, int out_size, void* d_ws, size_t ws_size,
                              hipStream_t stream) {
  (void)in_sizes; (void)n_in; (void)d_ws; (void)ws_size;
  const float* x = (const float*)d_in[0];   // (16, 64, 65536) fp32
  const float* w = (const float*)d_in[1];   // (64, 64, 512)   fp32
  float* out = (float*)d_out;               // (16, 64, 32769) fp32

  // 1) Zero the full output (store-bandwidth bound part).
  fill_zero_kernel<<<2048, 256, 0, stream>>>(out, (long long)out_size);

  // 2) Compute and write the first 512 spectral positions per (b, o).
  //    64 blocks * 8 waves/block = 512 waves, one per m.
  spectral_wmma_kernel<<<M_DIM / 8, 256, 0, stream>>>(x, w, out);
}